// DirectCostModule_80917183857256
// MI455X (gfx1250) — compile-verified
//
#include <hip/hip_runtime.h>
#include <cmath>

typedef float v2f __attribute__((ext_vector_type(2)));
typedef float v8f __attribute__((ext_vector_type(8)));

namespace {

constexpr int kV = 16;
constexpr int kN = 4096;
constexpr int kC = 128;
constexpr int kH = 480;
constexpr int kW = 640;
constexpr int kPad = 4;

// workspace layout (float offsets)
constexpr size_t WS_FPRN = 0;                                // N*C normalized ref features
constexpr size_t WS_MASK = WS_FPRN + (size_t)kN * kC;        // N
constexpr size_t WS_WR   = WS_MASK + kN;                     // N ref weights
constexpr size_t WS_P3DM = WS_WR + kN;                       // 3N masked points
constexpr size_t WS_K    = WS_P3DM + 3 * (size_t)kN;         // V*N*16 Gram rows

// output layout (float offsets, tuple order of reference)
constexpr size_t OUT_GRAD  = 0;                              // (V,6)
constexpr size_t OUT_HESS  = OUT_GRAD + (size_t)kV * 6;      // (V,6,6)
constexpr size_t OUT_WLOSS = OUT_HESS + (size_t)kV * 36;     // (V,N)
constexpr size_t OUT_VALQ  = OUT_WLOSS + (size_t)kV * kN;    // (V,N)
constexpr size_t OUT_P2DQ  = OUT_VALQ + (size_t)kV * kN;     // (V,N,2)
constexpr size_t OUT_COST  = OUT_P2DQ + (size_t)kV * kN * 2; // (V,N)

__device__ inline float clip2(float x) { return fminf(fmaxf(x, -2.f), 2.f); }

// bilinear grid_sample, align_corners=True, zero-padded border (reference semantics)
__device__ inline float bsample(const float* __restrict__ img, int c, float xn, float yn) {
  float x = (xn + 1.f) * 0.5f * (float)(kW - 1);
  float y = (yn + 1.f) * 0.5f * (float)(kH - 1);
  float x0f = floorf(x), y0f = floorf(y);
  float fx1 = x - x0f, fx0 = 1.f - fx1;
  float fy1 = y - y0f, fy0 = 1.f - fy1;
  int ix = (int)x0f, iy = (int)y0f;
  int px0 = min(max(ix + 1, 0), kW + 1);
  int px1 = min(max(ix + 2, 0), kW + 1);
  int py0 = min(max(iy + 1, 0), kH + 1);
  int py1 = min(max(iy + 2, 0), kH + 1);
  const float* base = img + (size_t)c * kW * kH;
  auto fetch = [&](int px, int py) -> float {
    bool in = (px >= 1) & (px <= kW) & (py >= 1) & (py <= kH);
    return in ? base[(size_t)(py - 1) * kW + (px - 1)] : 0.f;
  };
  return fetch(px0, py0) * (fx0 * fy0) + fetch(px0, py1) * (fx0 * fy1) +
         fetch(px1, py0) * (fx1 * fy0) + fetch(px1, py1) * (fx1 * fy1);
}

// ---------------- Kernel 1: reference view ----------------
__global__ void ref_pass(const float* __restrict__ f_r,
                         const float* __restrict__ c_ref,
                         const float* __restrict__ pose_r,
                         const float* __restrict__ cam_r,
                         const float* __restrict__ p3D,
                         float* __restrict__ ws) {
  int n = blockIdx.x;
  int c = threadIdx.x;
  float p0 = p3D[n * 3 + 0], p1 = p3D[n * 3 + 1], p2 = p3D[n * 3 + 2];
  float R0 = pose_r[0], R1 = pose_r[1], R2 = pose_r[2];
  float R3 = pose_r[3], R4 = pose_r[4], R5 = pose_r[5];
  float R6 = pose_r[6], R7 = pose_r[7], R8 = pose_r[8];
  float X = R0 * p0 + R1 * p1 + R2 * p2 + pose_r[9];
  float Y = R3 * p0 + R4 * p1 + R5 * p2 + pose_r[10];
  float Z = R6 * p0 + R7 * p1 + R8 * p2 + pose_r[11];
  float Wf = cam_r[0], Hf = cam_r[1];
  float fx = cam_r[2], fy = cam_r[3], cx = cam_r[4], cy = cam_r[5];
  float zc = fmaxf(Z, 1e-4f);
  float u = X / zc * fx + cx;
  float v = Y / zc * fy + cy;
  bool vis = (Z > 1e-4f) & (u >= 0.f) & (u <= Wf - 1.f) & (v >= 0.f) & (v <= Hf - 1.f);
  bool vr = (u >= (float)kPad) & (u <= Wf - (float)kPad - 1.f) &
            (v >= (float)kPad) & (v <= Hf - (float)kPad - 1.f);
  float m = (vis & vr) ? 1.f : 0.f;
  float xn = clip2(u / (float)(kW - 1) * 2.f - 1.f);
  float yn = clip2(v / (float)(kH - 1) * 2.f - 1.f);
  float fr = bsample(f_r, c, xn, yn);
  __shared__ float red[kC];
  red[c] = fr * fr;
  __syncthreads();
  for (int s = kC / 2; s > 0; s >>= 1) {
    if (c < s) red[c] += red[c + s];
    __syncthreads();
  }
  float nrm = sqrtf(red[0]);
  ws[WS_FPRN + (size_t)n * kC + c] = m * fr / fmaxf(nrm, 1e-12f);
  if (c == 0) {
    ws[WS_MASK + n] = m;
    ws[WS_WR + n] = bsample(c_ref, 0, xn, yn) * m;
    ws[WS_P3DM + 3 * n + 0] = p0 * m;
    ws[WS_P3DM + 3 * n + 1] = p1 * m;
    ws[WS_P3DM + 3 * n + 2] = p2 * m;
  }
}

// ---------------- Kernel 2: query views, residual+Jacobian row build ----------------
__global__ void query_pass(const float* __restrict__ f_q,
                           const float* __restrict__ c_query,
                           const float* __restrict__ pose_q,
                           const float* __restrict__ cam_q,
                           float* __restrict__ ws,
                           float* __restrict__ out) {
  int n = blockIdx.x;
  int v = blockIdx.y;
  int c = threadIdx.x;

  __builtin_prefetch(ws + WS_FPRN + (size_t)n * kC + c, 0, 0);  // global_prefetch_b8

  const float* P = pose_q + v * 12;
  float R0 = P[0], R1 = P[1], R2 = P[2];
  float R3 = P[3], R4 = P[4], R5 = P[5];
  float R6 = P[6], R7 = P[7], R8 = P[8];
  float p0 = ws[WS_P3DM + 3 * n + 0];
  float p1 = ws[WS_P3DM + 3 * n + 1];
  float p2 = ws[WS_P3DM + 3 * n + 2];
  float mask = ws[WS_MASK + n];
  float X = R0 * p0 + R1 * p1 + R2 * p2 + P[9];
  float Y = R3 * p0 + R4 * p1 + R5 * p2 + P[10];
  float Z = R6 * p0 + R7 * p1 + R8 * p2 + P[11];
  const float* cam = cam_q + v * 6;
  float Wf = cam[0], Hf = cam[1];
  float fxc = cam[2], fyc = cam[3], cx = cam[4], cy = cam[5];
  float zc = fmaxf(Z, 1e-4f);
  float u = X / zc * fxc + cx;
  float vv = Y / zc * fyc + cy;
  bool vis = (Z > 1e-4f) & (u >= 0.f) & (u <= Wf - 1.f) & (vv >= 0.f) & (vv <= Hf - 1.f);
  float um = u * mask, vm = vv * mask;
  bool vq = (um >= (float)kPad) & (um <= Wf - (float)kPad - 1.f) &
            (vm >= (float)kPad) & (vm <= Hf - (float)kPad - 1.f);
  float xn = clip2(um / (float)(kW - 1) * 2.f - 1.f);
  float yn = clip2(vm / (float)(kH - 1) * 2.f - 1.f);
  const float dxn = 2.f / (float)(kW - 1);
  const float dyn = 2.f / (float)(kH - 1);

  float f0  = bsample(f_q, c, xn, yn);
  float fxm = bsample(f_q, c, xn - dxn, yn);
  float fxp = bsample(f_q, c, xn + dxn, yn);
  float fym = bsample(f_q, c, xn, yn - dyn);
  float fyp = bsample(f_q, c, xn, yn + dyn);
  float res = f0 - ws[WS_FPRN + (size_t)n * kC + c];
  float gx = (fxp - fxm) * 0.5f;
  float gy = (fyp - fym) * 0.5f;

  __shared__ float red[6][kC];
  red[0][c] = res * res;
  red[1][c] = gx * res;
  red[2][c] = gy * res;
  red[3][c] = gx * gx;
  red[4][c] = gx * gy;
  red[5][c] = gy * gy;
  __syncthreads();
  for (int s = kC / 2; s > 0; s >>= 1) {
    if (c < s) {
#pragma unroll
      for (int j = 0; j < 6; j++) red[j][c] += red[j][c + s];
    }
    __syncthreads();
  }

  if (c == 0) {
    float s0 = red[0][0], ux = red[1][0], uy = red[2][0];
    float mxx = red[3][0], mxy = red[4][0], myy = red[5][0];

    // loss_fn1, truncate = 0.1
    float xl = s0 * 100.f;
    float cost = 2.f * logf(fminf(0.5f * xl, 3.3e38f) + 1.f) * 0.01f;
    float wl = 2.f / (xl + 2.f);
    float valq = (vq & vis) ? 1.f : 0.f;
    float wq = bsample(c_query, 0, xn, yn);
    float wgt = wl * valq * wq * ws[WS_WR + n];

    size_t vn = (size_t)v * kN + n;
    out[OUT_WLOSS + vn] = wl;
    out[OUT_VALQ + vn] = valq;
    out[OUT_P2DQ + 2 * vn + 0] = um;
    out[OUT_P2DQ + 2 * vn + 1] = vm;
    out[OUT_COST + vn] = cost;

    // A = fdiag * J_project(p3d_q) * [R*(-skew(p)) | R]   (2x6)
    float z2 = fmaxf(Z, 0.001f);
    float iz = 1.f / z2;
    float Rm[3][3] = {{R0, R1, R2}, {R3, R4, R5}, {R6, R7, R8}};
    float Sm[3][3] = {{0.f, p2, -p1}, {-p2, 0.f, p0}, {p1, -p0, 0.f}};
    float JP[3][6];
#pragma unroll
    for (int i = 0; i < 3; i++) {
#pragma unroll
      for (int k = 0; k < 3; k++) {
        float acc = 0.f;
#pragma unroll
        for (int j = 0; j < 3; j++) acc += Rm[i][j] * Sm[j][k];
        JP[i][k] = acc;
        JP[i][3 + k] = Rm[i][k];
      }
    }
    float r00 = fxc * iz, r02 = -fxc * X * iz * iz;
    float r11 = fyc * iz, r12 = -fyc * Y * iz * iz;
    float A0[6], A1[6];
#pragma unroll
    for (int f = 0; f < 6; f++) {
      A0[f] = r00 * JP[0][f] + r02 * JP[2][f];
      A1[f] = r11 * JP[1][f] + r12 * JP[2][f];
    }
    // Cholesky M = L^T L, L = [[a, b], [0, cc]];  rows: sqrt(w)*L*A, col6 = sqrt(w)*L^-T u
    float a = sqrtf(mxx + 1e-12f);
    float b = mxy / a;
    float cc = sqrtf(fmaxf(myy + 1e-12f - b * b, 1e-12f));
    float sw = sqrtf(fmaxf(wgt, 0.f));
    float q0 = sw * ux / a;
    float q1 = sw * (uy - b * ux / a) / cc;
    float* K = ws + WS_K + vn * 16;
#pragma unroll
    for (int f = 0; f < 6; f++) {
      K[f] = sw * (a * A0[f] + b * A1[f]);
      K[8 + f] = sw * cc * A1[f];
    }
    K[6] = q0;  K[7] = 0.f;
    K[14] = q1; K[15] = 0.f;
  }
}

// ---------------- Kernel 3: per-view Gram X^T X via WMMA f32 16x16x4 ----------------
// Branch-free operand build: every lane loads in-bounds (col&7) and pad columns
// (8..15) are zeroed with a multiply, so EXEC stays all-ones through the WMMA loop.
__global__ __launch_bounds__(32) void gram_pass(const float* __restrict__ ws,
                                                float* __restrict__ out) {
  int v = blockIdx.x;
  int lane = threadIdx.x;
  int col = lane & 15;
  int half = lane >> 4;
  int ci = col & 7;
  float msk = (col < 8) ? 1.f : 0.f;
  // lane-fixed base: row slice (half selects rows {0,1} vs {2,3} of the 4-row chunk)
  const float* pb = ws + WS_K + (size_t)v * kN * 16 + half * 16 + ci;

  v8f acc = {0.f, 0.f, 0.f, 0.f, 0.f, 0.f, 0.f, 0.f};
  for (int i = 0; i < kN / 2; i++) {
    v2f a;
    a.x = pb[0] * msk;  // A VGPR0: K rows 0 / 2
    a.y = pb[8] * msk;  // A VGPR1: K rows 1 / 3
    // D = A * B + C ; Gram: same operand as A and B (symmetric layouts)
    acc = __builtin_amdgcn_wmma_f32_16x16x4_f32(
        /*neg_a=*/false, a, /*neg_b=*/false, a,
        /*c_mod=*/(short)0, acc, /*reuse_a=*/false, /*reuse_b=*/false);
    pb += 32;  // next 2 points (4 rows)
  }
  // D[f][g] lives at lane g (g<16), VGPR f (f<8). H = D[0:6][0:6], grad = D[0:6][6].
  if (lane < 6) {
#pragma unroll
    for (int f = 0; f < 6; f++)
      out[OUT_HESS + (size_t)v * 36 + f * 6 + lane] = acc[f];
    out[OUT_GRAD + (size_t)v * 6 + lane] = -acc[6];
  }
}

}  // namespace

extern "C" void kernel_launch(void* const* d_in, const int* in_sizes, int n_in,
                              void* d_out, int out_size, void* d_ws, size_t ws_size,
                              hipStream_t stream) {
  const float* pose_q = (const float*)d_in[0];
  const float* f_r    = (const float*)d_in[1];
  const float* pose_r = (const float*)d_in[2];
  const float* cam_r  = (const float*)d_in[3];
  const float* f_q    = (const float*)d_in[4];
  const float* cam_q  = (const float*)d_in[5];
  const float* p3D    = (const float*)d_in[6];
  const float* c_ref  = (const float*)d_in[7];
  const float* c_qry  = (const float*)d_in[8];
  float* out = (float*)d_out;
  float* ws  = (float*)d_ws;

  ref_pass<<<dim3(kN), dim3(kC), 0, stream>>>(f_r, c_ref, pose_r, cam_r, p3D, ws);
  query_pass<<<dim3(kN, kV), dim3(kC), 0, stream>>>(f_q, c_qry, pose_q, cam_q, ws, out);
  gram_pass<<<dim3(kV), dim3(32), 0, stream>>>(ws, out);
}